// NCC_84791244357756
// MI455X (gfx1250) — compile-verified
//
#include <hip/hip_runtime.h>
#include <math.h>

// ---------------------------------------------------------------------------
// NCC on MI455X (gfx1250, wave32).
// Main conv as implicit GEMM on V_WMMA_F32_16X16X4_F32:
//   per image row r:  R[x(16), dy(32)] = A(16x32 Hankel of row) * B(32x32 wts)
//   B col 31 = uniform 1/961 -> box row-sum for local mean (m1)
//   out[y,x] = sum_dy R[y-15+dy, x, dy]  via 40-slot LDS ring
// m2 (mean of squares) via 31-tap VALU row sums in a parallel ring.
// Ring slots tracked incrementally (no div/mod in the hot loop).
// ---------------------------------------------------------------------------

typedef __attribute__((ext_vector_type(2))) float v2f;
typedef __attribute__((ext_vector_type(8))) float v8f;

#define IMG_H 1024
#define IMG_W 1024
#define NB    8
#define NC    3
#define TDIM  31
#define PADR  15
#define PE    961.0f

#define XT    16          // output x per block
#define YS    128         // output rows per block (strip)
#define RING  40          // R-row ring slots (>= 38 needed by pipeline)
#define PADX  17          // x stride in ring (bank-conflict-free)

__device__ __forceinline__ int wrapRing(int s) { return (s >= RING) ? s - RING : s; }

// --------------------------- template prep ---------------------------------
// wpad[c][dy][dx], dy 0..31 (31 = uniform 1/961 box column), dx 0..31 (31 = 0)
__global__ void ncc_prep_kernel(const float* __restrict__ tmpl,
                                float* __restrict__ wpad) {
  const int c = blockIdx.x;
  const float* t = tmpl + c * (TDIM * TDIM);
  __shared__ float red[256], red2[256];
  float s = 0.f, s2 = 0.f;
  for (int i = threadIdx.x; i < TDIM * TDIM; i += 256) {
    float v = t[i]; s += v; s2 += v * v;
  }
  red[threadIdx.x] = s; red2[threadIdx.x] = s2;
  __syncthreads();
  for (int off = 128; off > 0; off >>= 1) {
    if (threadIdx.x < off) {
      red[threadIdx.x]  += red[threadIdx.x + off];
      red2[threadIdx.x] += red2[threadIdx.x + off];
    }
    __syncthreads();
  }
  const float mu  = red[0] / PE;
  const float var = fmaxf(red2[0] / PE - mu * mu, 0.f);
  const float sd  = sqrtf(var);
  const float inv = 1.0f / ((sd + 1e-9f) * PE);
  for (int i = threadIdx.x; i < 32 * 32; i += 256) {
    const int dy = i >> 5, dx = i & 31;
    float v = 0.f;
    if (dx < TDIM) {
      v = (dy < TDIM) ? (t[dy * TDIM + dx] - mu) * inv : (1.0f / PE);
    }
    wpad[c * 1024 + i] = v;
  }
}

// ----------------------------- main kernel ---------------------------------
__global__ __launch_bounds__(256)
void ncc_main_kernel(const float* __restrict__ img,
                     const float* __restrict__ wpad,
                     float* __restrict__ out) {
  __shared__ float ringR[RING * 32 * PADX];  // [slot][dy 0..31][x (pad 17)]
  __shared__ float ring2[RING * XT];         // box row-sums of img^2 / 961
  __shared__ float accs[YS * XT];            // cross-channel accumulator
  __shared__ float wbuf[32 * 32];            // staged B matrix
  __shared__ float rowbuf[8][48];            // per-wave image-row segment

  const int tid  = threadIdx.x;
  const int wave = tid >> 5;
  const int lane = tid & 31;
  const int x0 = blockIdx.x * XT;
  const int y0 = blockIdx.y * YS;
  const int b  = blockIdx.z;
  const int xbase = x0 - PADR;

  const int m     = lane & 15;          // A row / B col within half
  const int koff  = (lane >> 4) << 1;   // K sub-offset: 0 (lanes<16) or 2
  const int xo    = (lane >> 4) << 3;   // D: M offset 0/8
  const int dy0   = lane & 15;          // D: N index

  // one-time ring-slot seeds (only mod in the kernel)
  const int slotW0 = (((y0 - PADR + wave) % RING) + RING) % RING;          // stage 1
  const int slotE0 = (((y0 - PADR + (tid >> 4)) % RING) + RING) % RING;    // stage 2

  for (int c = 0; c < NC; ++c) {
    // stage normalized weights for this channel
    for (int i = tid; i < 32 * 32; i += 256) wbuf[i] = wpad[c * 1024 + i];
    __syncthreads();

    // loop-invariant B fragments: 8 K-chunks x 2 N-tiles
    v2f Bfrag[8][2];
#pragma unroll
    for (int k = 0; k < 8; ++k) {
#pragma unroll
      for (int nt = 0; nt < 2; ++nt) {
        const int dy = m + (nt << 4);
        const int dx = (k << 2) + koff;
        v2f bv; bv.x = wbuf[dy * 32 + dx]; bv.y = wbuf[dy * 32 + dx + 1];
        Bfrag[k][nt] = bv;
      }
    }

    const float* imgc = img + ((size_t)b * NC + c) * (size_t)(IMG_H * IMG_W);

    int slotW = slotW0;   // stage-1 ring slot, advances +8 per group
    int slotE = slotE0;   // stage-2 ring slot, advances +8 per emit group

    const int ngroups = 4 + YS / 8;  // 4-group pipeline depth + 16 emit groups
    for (int g = 0; g < ngroups; ++g) {
      // ================= stage 1: each wave computes one R row =============
      const int irow = y0 - PADR + 8 * g + wave;  // image row index
      {
        const bool rowok = (irow >= 0) && (irow < IMG_H);
        const float* rp = imgc + (size_t)(rowok ? irow : 0) * IMG_W;
        for (int j = lane; j < 48; j += 32) {
          const int col = xbase + j;
          const bool ok = rowok && (col >= 0) && (col < IMG_W) && (j < 46);
          rowbuf[wave][j] = ok ? rp[col] : 0.f;
        }
        // speculative prefetch of this wave's row 8 groups ahead
        {
          int pr = irow + 64; if (pr >= IMG_H) pr = IMG_H - 1;
          int pc = xbase < 0 ? 0 : xbase;
          __builtin_prefetch(imgc + (size_t)pr * IMG_W + pc, 0, 0);
        }

        v8f c0 = {}; v8f c1 = {};
#pragma unroll
        for (int k = 0; k < 8; ++k) {
          const int base = m + (k << 2) + koff;
          v2f a; a.x = rowbuf[wave][base]; a.y = rowbuf[wave][base + 1];
          c0 = __builtin_amdgcn_wmma_f32_16x16x4_f32(
              false, a, false, Bfrag[k][0], (short)0, c0, false, false);
          c1 = __builtin_amdgcn_wmma_f32_16x16x4_f32(
              false, a, false, Bfrag[k][1], (short)0, c1, false, false);
        }

        float* rs = &ringR[slotW * 32 * PADX];
#pragma unroll
        for (int v = 0; v < 8; ++v) {
          rs[dy0 * PADX + xo + v]        = c0[v];   // dy 0..15
          rs[(dy0 + 16) * PADX + xo + v] = c1[v];   // dy 16..31 (31 = box)
        }
        if (lane < XT) {                            // m2 row sums
          float s = 0.f;
#pragma unroll 1
          for (int dxx = 0; dxx < TDIM; ++dxx) {
            const float vv = rowbuf[wave][lane + dxx];
            s = fmaf(vv, vv, s);
          }
          ring2[slotW * XT + lane] = s * (1.0f / PE);
        }
        slotW = wrapRing(slotW + 8);
      }
      __syncthreads();

      // ================= stage 2: emit 8 output rows =======================
      if (g >= 4 && tid < 128) {
        const int e  = g - 4;
        const int yy = 8 * e + (tid >> 4);   // row within strip
        const int x  = tid & 15;
        const int y  = y0 + yy;
        int slot = slotE;                    // slot of row y-15
        slotE = wrapRing(slotE + 8);
        float conv = 0.f, m1 = 0.f, m2 = 0.f;
#pragma unroll 1
        for (int dy = 0; dy < TDIM; ++dy) {
          const float* rs = &ringR[slot * 32 * PADX];
          conv += rs[dy * PADX + x];
          m1   += rs[31 * PADX + x];
          m2   += ring2[slot * XT + x];
          slot = wrapRing(slot + 1);
        }
        const float sd  = sqrtf(m2 - m1 * m1);  // NaN if var<0, as reference
        const float val = conv / (sd + 1e-9f);
        const int ai = yy * XT + x;
        float a = (c == 0) ? val : (accs[ai] + val);
        accs[ai] = a;
        if (c == NC - 1) {
          float o = a * (1.0f / 3.0f);
          if (o != o) o = 0.f;                 // NaN -> 0 after channel mean
          out[((size_t)b * IMG_H + y) * IMG_W + (x0 + x)] = o;
        }
      }
      __syncthreads();  // protect ring rows still read above from next writes
    }
  }
}

// ----------------------------- launcher ------------------------------------
extern "C" void kernel_launch(void* const* d_in, const int* in_sizes, int n_in,
                              void* d_out, int out_size, void* d_ws, size_t ws_size,
                              hipStream_t stream) {
  const float* img  = (const float*)d_in[0];   // (8,3,1024,1024) f32
  const float* tmpl = (const float*)d_in[1];   // (3,31,31) f32
  float* wpad = (float*)d_ws;                  // 3*32*32 f32 = 12 KB
  float* outp = (float*)d_out;                 // (8,1024,1024) f32

  ncc_prep_kernel<<<dim3(NC), dim3(256), 0, stream>>>(tmpl, wpad);

  dim3 grid(IMG_W / XT, IMG_H / YS, NB);       // (64, 8, 8)
  ncc_main_kernel<<<grid, dim3(256), 0, stream>>>(img, wpad, outp);
}